// RFAGNN_16956530884762
// MI455X (gfx1250) — compile-verified
//
#include <hip/hip_runtime.h>
#include <math.h>

#define IN_DIM   500
#define GDIM     64
#define HIDDEN   64
#define OUT_DIM  7
#define BETA_F   0.3f

typedef float v2f __attribute__((ext_vector_type(2)));
typedef float v8f __attribute__((ext_vector_type(8)));

// ---------------------------------------------------------------- utilities
__global__ void rfagnn_zero_kernel(float* __restrict__ p, long long n) {
  long long i = (long long)blockIdx.x * blockDim.x + threadIdx.x;
  if (i < n) p[i] = 0.0f;
}

__global__ void rfagnn_deg_kernel(const int* __restrict__ dst,
                                  float* __restrict__ deg, int E) {
  int e = blockIdx.x * blockDim.x + threadIdx.x;
  if (e < E) atomicAdd(deg + dst[e], 1.0f);
}

__global__ void rfagnn_dinv_kernel(float* __restrict__ d, int n) {
  int i = blockIdx.x * blockDim.x + threadIdx.x;
  if (i < n) d[i] = rsqrtf(fmaxf(d[i], 1.0f));
}

// ------------------------------------------------- fused dual GEMM via WMMA
// h  = x @ Wl + bl                (graph-learning projection)
// h2 = relu(x @ W1 + b1)          (trunk projection)
// One block = 16 rows of x, 8 waves: wave/4 selects matrix, wave%4 the
// 16-wide column tile. K loop uses V_WMMA_F32_16X16X4_F32 (full fp32).
__global__ __launch_bounds__(256)
void rfagnn_dual_gemm_kernel(const float* __restrict__ x,
                             const float* __restrict__ Wl, const float* __restrict__ bl,
                             const float* __restrict__ W1, const float* __restrict__ b1,
                             float* __restrict__ h, float* __restrict__ h2) {
  const int lane  = threadIdx.x & 31;
  const int wave  = threadIdx.x >> 5;
  const int half  = lane >> 4;        // 0: K pair {0,1}, 1: K pair {2,3}
  const int l     = lane & 15;        // M index (A) / N index (B, C)
  const int mat   = wave >> 2;        // 0 -> Wl/h, 1 -> W1/h2(relu)
  const int ctile = (wave & 3) << 4;
  const int rowBase = blockIdx.x << 4;

  const float* __restrict__ W    = mat ? W1 : Wl;
  const float* __restrict__ bias = mat ? b1 : bl;
  float* __restrict__ outp       = mat ? h2 : h;

  // A: lane holds x[rowBase+l][k0 + 2*half + {0,1}]  (8-byte aligned float2)
  const float* ap = x + (size_t)(rowBase + l) * IN_DIM + 2 * half;
  // B: lane holds W[(k0 + 2*half + {0,1})][ctile + l]
  const float* bp = W + (size_t)(2 * half) * GDIM + ctile + l;

  v8f acc = {};
#pragma unroll 5
  for (int k0 = 0; k0 < IN_DIM; k0 += 4) {
    v2f a = *(const v2f*)(ap + k0);
    v2f b;
    b.x = bp[(size_t)k0 * GDIM];
    b.y = bp[(size_t)k0 * GDIM + GDIM];
    acc = __builtin_amdgcn_wmma_f32_16x16x4_f32(
        /*neg_a=*/false, a, /*neg_b=*/false, b,
        /*c_mod=*/(short)0, acc, /*reuse_a=*/false, /*reuse_b=*/false);
  }

  // C/D layout: lane<16 -> rows 0..7, lane>=16 -> rows 8..15, col = l+ctile
  const int col = ctile + l;
  const float bb = bias[col];
  float* orow = outp + (size_t)(rowBase + half * 8) * GDIM + col;
#pragma unroll
  for (int r = 0; r < 8; ++r) {
    float v = acc[r] + bb;
    if (mat) v = fmaxf(v, 0.0f);
    orow[(size_t)r * GDIM] = v;
  }
}

// ----------------------------------------- node-level gate projections
// pd[n] = feat[n] . w[0:64] ; ps[n] = feat[n] . w[64:128]
__global__ __launch_bounds__(256)
void rfagnn_gate_proj_kernel(const float* __restrict__ feat,
                             const float* __restrict__ w,
                             float* __restrict__ pd, float* __restrict__ ps, int N) {
  __shared__ float sw[2 * HIDDEN];
  if (threadIdx.x < 2 * HIDDEN) sw[threadIdx.x] = w[threadIdx.x];
  __syncthreads();
  int i = blockIdx.x * blockDim.x + threadIdx.x;
  if (i >= N) return;
  const float* f = feat + (size_t)i * HIDDEN;
  float accd = 0.0f, accs = 0.0f;
#pragma unroll
  for (int k = 0; k < HIDDEN; ++k) {
    float v = f[k];
    accd = fmaf(v, sw[k], accd);
    accs = fmaf(v, sw[HIDDEN + k], accs);
  }
  pd[i] = accd;
  ps[i] = accs;
}

// ----------------------------------------- edge gather/scale/scatter-add
// 16 lanes per edge, float4 per lane. feat/agg are L2-resident (12.8 MB).
__global__ __launch_bounds__(256)
void rfagnn_edge_msg_kernel(const int* __restrict__ src, const int* __restrict__ dst,
                            const float* __restrict__ feat,
                            const float* __restrict__ pd, const float* __restrict__ ps,
                            const float* __restrict__ bias, int bias_idx,
                            const float* __restrict__ d,
                            float* __restrict__ agg, int E) {
  long long gid = (long long)blockIdx.x * blockDim.x + threadIdx.x;
  long long e = gid >> 4;
  if (e >= E) return;
  const int sub = (int)(gid & 15);
  const int s = src[e], t = dst[e];
  const float fct = tanhf(pd[t] + ps[s] + bias[bias_idx]) * d[t] * d[s];
  const float4 v = *(const float4*)(feat + (size_t)s * HIDDEN + sub * 4);
  float* ag = agg + (size_t)t * HIDDEN + sub * 4;
  atomicAdd(ag + 0, v.x * fct);
  atomicAdd(ag + 1, v.y * fct);
  atomicAdd(ag + 2, v.z * fct);
  atomicAdd(ag + 3, v.w * fct);
}

// feat = agg*(1-BETA) + h2*BETA
__global__ void rfagnn_feat_update_kernel(float* __restrict__ feat,
                                          const float* __restrict__ agg,
                                          const float* __restrict__ h2, long long n) {
  long long i = (long long)blockIdx.x * blockDim.x + threadIdx.x;
  if (i < n) feat[i] = agg[i] * (1.0f - BETA_F) + h2[i] * BETA_F;
}

// out[n] (g==0: init with b2) += feat_g[n] @ W2[g*64:(g+1)*64, :]
__global__ __launch_bounds__(256)
void rfagnn_final_linear_kernel(const float* __restrict__ feat,
                                const float* __restrict__ W2,
                                const float* __restrict__ b2,
                                float* __restrict__ out, int g, int N) {
  __shared__ float sw[HIDDEN * OUT_DIM];
  for (int i = threadIdx.x; i < HIDDEN * OUT_DIM; i += blockDim.x)
    sw[i] = W2[(size_t)(g * HIDDEN) * OUT_DIM + i];
  __syncthreads();
  int n = blockIdx.x * blockDim.x + threadIdx.x;
  if (n >= N) return;
  const float* f = feat + (size_t)n * HIDDEN;
  float acc[OUT_DIM];
#pragma unroll
  for (int j = 0; j < OUT_DIM; ++j)
    acc[j] = (g == 0) ? b2[j] : out[(size_t)n * OUT_DIM + j];
  for (int k = 0; k < HIDDEN; ++k) {
    float v = f[k];
#pragma unroll
    for (int j = 0; j < OUT_DIM; ++j) acc[j] = fmaf(v, sw[k * OUT_DIM + j], acc[j]);
  }
#pragma unroll
  for (int j = 0; j < OUT_DIM; ++j) out[(size_t)n * OUT_DIM + j] = acc[j];
}

// ---------------------------------------------------------------- launcher
extern "C" void kernel_launch(void* const* d_in, const int* in_sizes, int n_in,
                              void* d_out, int out_size, void* d_ws, size_t ws_size,
                              hipStream_t stream) {
  const float* x   = (const float*)d_in[0];
  const int*   src = (const int*)d_in[1];
  const int*   dst = (const int*)d_in[2];
  const float* Wl  = (const float*)d_in[3];
  const float* bl  = (const float*)d_in[4];
  const float* glw = (const float*)d_in[5];   // [2, 128]
  const float* glb = (const float*)d_in[6];   // [2]
  const float* W1  = (const float*)d_in[7];
  const float* b1  = (const float*)d_in[8];
  const float* faw = (const float*)d_in[9];   // [2, 2, 128]
  const float* fab = (const float*)d_in[10];  // [2, 2]
  const float* W2  = (const float*)d_in[11];  // [128, 7]
  const float* b2  = (const float*)d_in[12];  // [7]
  float* out = (float*)d_out;

  const int N = in_sizes[0] / IN_DIM;   // 50000 (divisible by 16)
  const int E = in_sizes[1];            // 1600000

  // workspace carve-out (floats)
  float* p = (float*)d_ws;
  float* d    = p; p += N;
  float* h    = p; p += (size_t)N * GDIM;
  float* h2   = p; p += (size_t)N * HIDDEN;
  float* feat = p; p += (size_t)N * HIDDEN;
  float* agg  = p; p += (size_t)N * HIDDEN;
  float* pdg0 = p; p += N;
  float* psg0 = p; p += N;
  float* pdg1 = p; p += N;
  float* psg1 = p; p += N;
  float* pdt  = p; p += N;
  float* pst  = p; p += N;

  const dim3 b256(256);
  auto nblk = [](long long n) { return dim3((unsigned)((n + 255) / 256)); };
  const long long nh = (long long)N * HIDDEN;

  // degree -> d = clip(deg,1)^-0.5
  rfagnn_zero_kernel<<<nblk(N), b256, 0, stream>>>(d, N);
  rfagnn_deg_kernel<<<nblk(E), b256, 0, stream>>>(dst, d, E);
  rfagnn_dinv_kernel<<<nblk(N), b256, 0, stream>>>(d, N);

  // fused dual GEMM: h = x@Wl+bl ; h2 = relu(x@W1+b1)   (WMMA f32)
  rfagnn_dual_gemm_kernel<<<dim3(N / 16), b256, 0, stream>>>(x, Wl, bl, W1, b1, h, h2);

  // GraphLearning gate projections from h (hop-0 factors, both graphs)
  rfagnn_gate_proj_kernel<<<nblk(N), b256, 0, stream>>>(h, glw,             pdg0, psg0, N);
  rfagnn_gate_proj_kernel<<<nblk(N), b256, 0, stream>>>(h, glw + 2 * GDIM,  pdg1, psg1, N);

  for (int g = 0; g < 2; ++g) {
    const float* pdg = g ? pdg1 : pdg0;
    const float* psg = g ? psg1 : psg0;

    // hop 0: feat_in = h2, gate = GraphLearning factor
    rfagnn_zero_kernel<<<nblk(nh), b256, 0, stream>>>(agg, nh);
    rfagnn_edge_msg_kernel<<<nblk((long long)E * 16), b256, 0, stream>>>(
        src, dst, h2, pdg, psg, glb, g, d, agg, E);
    rfagnn_feat_update_kernel<<<nblk(nh), b256, 0, stream>>>(feat, agg, h2, nh);

    // hop 1: gate from FALayer weights fa_w[g,1], feat_in = feat
    rfagnn_gate_proj_kernel<<<nblk(N), b256, 0, stream>>>(
        feat, faw + (size_t)(g * 2 + 1) * 2 * HIDDEN, pdt, pst, N);
    rfagnn_zero_kernel<<<nblk(nh), b256, 0, stream>>>(agg, nh);
    rfagnn_edge_msg_kernel<<<nblk((long long)E * 16), b256, 0, stream>>>(
        src, dst, feat, pdt, pst, fab, g * 2 + 1, d, agg, E);
    rfagnn_feat_update_kernel<<<nblk(nh), b256, 0, stream>>>(feat, agg, h2, nh);

    // out (+)= feat @ W2[g*64:(g+1)*64, :]  (g==0 initializes with b2)
    rfagnn_final_linear_kernel<<<nblk(N), b256, 0, stream>>>(feat, W2, b2, out, g, N);
  }
}